// TI_Loss_11819749998724
// MI455X (gfx1250) — compile-verified
//
#include <hip/hip_runtime.h>
#include <hip/hip_bf16.h>

#define VOL (128*128*128)   // 2,097,152 voxels per batch volume
#define BATCH 2
#define TY 8                // y-tile
#define TZ 4                // z-tile
#define HY (TY+2)           // 10 halo rows in y
#define HZ (TZ+2)           // 6 halo slices in z
#define HROWS (HY*HZ)       // 60 halo rows of 128 bytes
#define NPART (BATCH*(128/TY)*(128/TZ))   // 1024 blocks in pass 2

typedef __attribute__((ext_vector_type(2))) float v2f;
typedef __attribute__((ext_vector_type(4))) float v4f;
typedef __attribute__((ext_vector_type(8))) float v8f;

// ---- gfx1250 async global->LDS copy (ASYNCcnt path) -----------------------
// VDST = LDS byte offset (low 32 bits of flat shared pointer == LDS offset on
// gfx1250: LDS aperture lives in addr[63:32]). GV mode: vaddr = 64-bit global.
__device__ __forceinline__ void async_ld_b128(const void* g, unsigned int lds_off) {
  asm volatile("global_load_async_to_lds_b128 %0, %1, off"
               :: "v"(lds_off), "v"(g) : "memory");
}
__device__ __forceinline__ void async_wait0() {
  asm volatile("s_wait_asynccnt 0" ::: "memory");
}

// exact per-byte zero detector (no cross-byte borrow false positives)
__device__ __forceinline__ unsigned int zbytes(unsigned int t) {
  unsigned int m = (t & 0x7F7F7F7Fu) + 0x7F7F7F7Fu;
  return ~(m | t) & 0x80808080u;
}
// gather the four 0x80 flags into a nibble (byte k -> bit k)
__device__ __forceinline__ unsigned int nib(unsigned int z) {
  return ((z >> 7) & 1u) | ((z >> 14) & 2u) | ((z >> 21) & 4u) | ((z >> 28) & 8u);
}

// ---------------- Pass 1: fused argmax + cross-entropy ---------------------
// one thread = 4 consecutive voxels; x read once (nontemporal), P(u8)+CE(f32) out
__global__ __launch_bounds__(256) void k_argmax_ce(
    const float* __restrict__ x, const int* __restrict__ y,
    unsigned char* __restrict__ P, float* __restrict__ ce)
{
  int gid = blockIdx.x * 256 + threadIdx.x;         // quad index over both batches
  long idx4 = (long)gid * 4;
  int  b = (int)(idx4 >> 21);                       // / VOL
  long v = idx4 & (long)(VOL - 1);

  const v4f* x4 = (const v4f*)x;
  long xq = (long)b * VOL + (v >> 2);               // quad index of class 0
  v4f xv[4];
#pragma unroll
  for (int c = 0; c < 4; ++c)
    xv[c] = __builtin_nontemporal_load(&x4[xq + (long)c * (VOL / 4)]);
  int4 yv = ((const int4*)y)[gid];

  uchar4 pv; v4f cev;
  unsigned char* pp = (unsigned char*)&pv;
#pragma unroll
  for (int j = 0; j < 4; ++j) {
    float a0 = xv[0][j];
    float a1 = xv[1][j];
    float a2 = xv[2][j];
    float a3 = xv[3][j];
    int best = 0; float bm = a0;                     // first-max == jnp.argmax
    if (a1 > bm) { bm = a1; best = 1; }
    if (a2 > bm) { bm = a2; best = 2; }
    if (a3 > bm) { bm = a3; best = 3; }
    float s = expf(a0 - bm) + expf(a1 - bm) + expf(a2 - bm) + expf(a3 - bm);
    float lse = bm + logf(s);
    int lbl = ((const int*)&yv)[j] & 3;
    float ay = (lbl == 0) ? a0 : (lbl == 1) ? a1 : (lbl == 2) ? a2 : a3;
    cev[j] = lse - ay;
    pp[j] = (unsigned char)best;
  }
  ((uchar4*)P)[gid] = pv;
  ((v4f*)ce)[gid] = cev;
}

// ---------------- Pass 2: critical voxels (SWAR stencil) + masked sum ------
__global__ __launch_bounds__(256) void k_crit_sum(
    const unsigned char* __restrict__ P, const float* __restrict__ ce,
    float* __restrict__ partial)
{
  __shared__ __align__(16) unsigned char sL[HROWS * 128];     // halo label rows
  __shared__ unsigned int sM1[HROWS][4], sM03[HROWS][4];      // per-row bitmasks
  __shared__ unsigned int sD1[TY * TZ][4], sD03[TY * TZ][4];  // 9-row vertical OR
  __shared__ float sred[256];

  int t = threadIdx.x;
  int bid = blockIdx.x;
  int tz = bid & 31, ty = (bid >> 5) & 15, b = bid >> 9;
  int z0 = tz * TZ, y0 = ty * TY;
  const unsigned char* Pb = P + (long)b * VOL;

  // Stage 0: async-copy valid halo rows (128B each, 8 x B128) into LDS
  for (int i = t; i < HROWS * 8; i += 256) {
    int row = i >> 3, ck = i & 7;
    int lz = row / HY, ly = row - lz * HY;
    int z = z0 - 1 + lz, y = y0 - 1 + ly;
    if ((unsigned)z < 128u && (unsigned)y < 128u) {
      const unsigned char* g = Pb + (((long)z * 128 + y) * 128) + ck * 16;
      unsigned int lo = (unsigned int)(unsigned long long)&sL[row * 128 + ck * 16];
      async_ld_b128(g, lo);
    }
  }
  async_wait0();
  __syncthreads();

  // Stage 1: per-row 128-bit masks: m1 = (label==1), m03 = (label in {0,3})
  for (int i = t; i < HROWS * 4; i += 256) {               // 240 word tasks
    int row = i >> 2, w = i & 3;
    int lz = row / HY, ly = row - lz * HY;
    int z = z0 - 1 + lz, y = y0 - 1 + ly;
    unsigned int m1 = 0, m03 = 0;
    if ((unsigned)z < 128u && (unsigned)y < 128u) {        // OOB rows => zero mask
      const unsigned int* rp = (const unsigned int*)&sL[row * 128 + w * 32];
#pragma unroll
      for (int k = 0; k < 8; ++k) {
        unsigned int vv = rp[k];
        m1  |= nib(zbytes(vv ^ 0x01010101u)) << (k * 4);
        m03 |= (nib(zbytes(vv)) | nib(zbytes(vv ^ 0x03030303u))) << (k * 4);
      }
    }
    sM1[row][w] = m1; sM03[row][w] = m03;
  }
  __syncthreads();

  // Stage 2: OR the 9 neighbor rows (y,z dilation); 256 tasks = 32 rows x 4w x 2
  {
    int msel = t & 1, w = (t >> 1) & 3, r2 = t >> 3;       // r2 = oz*TY + oy
    int oz = r2 >> 3, oy = r2 & 7;
    unsigned int acc = 0;
#pragma unroll
    for (int dz = 0; dz < 3; ++dz)
#pragma unroll
      for (int dy = 0; dy < 3; ++dy) {
        int row = (oz + dz) * HY + (oy + dy);
        acc |= msel ? sM03[row][w] : sM1[row][w];
      }
    if (msel) sD03[r2][w] = acc; else sD1[r2][w] = acc;
  }
  __syncthreads();

  // Stage 3: x-dilation by shifts, select per-label rule, accumulate CE
  float sum = 0.f;
  if (t < 128) {                                           // 32 rows x 4 words
    int w = t & 3, r2 = t >> 2;
    int oz = r2 >> 3, oy = r2 & 7;
    int z = z0 + oz, y = y0 + oy;
    unsigned int c1 = sD1[r2][w],  p1 = w ? sD1[r2][w - 1] : 0u,  n1 = (w < 3) ? sD1[r2][w + 1] : 0u;
    unsigned int c3 = sD03[r2][w], p3 = w ? sD03[r2][w - 1] : 0u, n3 = (w < 3) ? sD03[r2][w + 1] : 0u;
    unsigned int D1  = c1 | (c1 << 1) | (c1 >> 1) | (p1 >> 31) | (n1 << 31);
    unsigned int D03 = c3 | (c3 << 1) | (c3 >> 1) | (p3 >> 31) | (n3 << 31);
    int rowc = (oz + 1) * HY + (oy + 1);                   // center row (always valid)
    const unsigned int* rp = (const unsigned int*)&sL[rowc * 128 + w * 32];
    const v4f* ce4 = (const v4f*)(ce + (((long)b * 128 + z) * 128 + y) * 128 + w * 32);
#pragma unroll
    for (int k = 0; k < 8; ++k) {
      unsigned int lv = rp[k];
      v4f cv = ce4[k];
#pragma unroll
      for (int j = 0; j < 4; ++j) {
        unsigned int p = (lv >> (8 * j)) & 255u;
        int bit = k * 4 + j;
        unsigned int crit = (p == 1u) ? ((D03 >> bit) & 1u)
                          : ((p == 0u) | (p == 3u)) ? ((D1 >> bit) & 1u) : 0u;
        if (crit) sum += cv[j];
      }
    }
  }

  sred[t] = sum;
  __syncthreads();
  for (int s = 128; s > 0; s >>= 1) {
    if (t < s) sred[t] += sred[t + s];
    __syncthreads();
  }
  if (t == 0) partial[bid] = sred[0];
}

// ---------------- Pass 3: single-wave WMMA reduction of 1024 partials ------
// A = 16x4 ones => D[m][n] = sum_k B[k][n] + C[m][n]; every D row identical,
// so sum(c[0] over 32 lanes) = 2 * sum(B). Scale: 1/2 (row dup) * 1/2 (batch mean).
__global__ __launch_bounds__(32) void k_final(const float* __restrict__ partial,
                                              float* __restrict__ out)
{
  int lane = threadIdx.x;
  __shared__ float sv[32];
#if __has_builtin(__builtin_amdgcn_wmma_f32_16x16x4_f32)
  v2f a; a.x = 1.0f; a.y = 1.0f;
  v8f c = {};
  for (int i = 0; i < NPART; i += 64) {
    v2f bvec;
    bvec.x = partial[i + lane * 2 + 0];
    bvec.y = partial[i + lane * 2 + 1];
    c = __builtin_amdgcn_wmma_f32_16x16x4_f32(false, a, false, bvec,
                                              (short)0, c, false, false);
  }
  sv[lane] = c[0];
  __syncthreads();
  if (lane == 0) {
    float s = 0.f;
    for (int i = 0; i < 32; ++i) s += sv[i];
    out[0] = s * 0.25f;
  }
#else
  float s = 0.f;
  for (int i = lane; i < NPART; i += 32) s += partial[i];
  sv[lane] = s;
  __syncthreads();
  if (lane == 0) {
    float tot = 0.f;
    for (int i = 0; i < 32; ++i) tot += sv[i];
    out[0] = tot * 0.5f;
  }
#endif
}

// ---------------------------------------------------------------------------
extern "C" void kernel_launch(void* const* d_in, const int* in_sizes, int n_in,
                              void* d_out, int out_size, void* d_ws, size_t ws_size,
                              hipStream_t stream)
{
  const float* x = (const float*)d_in[0];   // (2,4,128,128,128) f32
  const int*   y = (const int*)d_in[1];     // (2,1,128,128,128) int

  // workspace layout (~21 MB): P u8 [4 MiB] | CE f32 [16 MiB] | partials [4 KiB]
  unsigned char* P  = (unsigned char*)d_ws;
  float* ce      = (float*)((char*)d_ws + (size_t)BATCH * VOL);
  float* partial = (float*)((char*)d_ws + (size_t)BATCH * VOL + (size_t)BATCH * VOL * 4);
  float* out = (float*)d_out;

  k_argmax_ce<<<(BATCH * VOL) / (256 * 4), 256, 0, stream>>>(x, y, P, ce);
  k_crit_sum<<<NPART, 256, 0, stream>>>(P, ce, partial);
  k_final<<<1, 32, 0, stream>>>(partial, out);
}